// SecondOrderFeatureInteraction_6889127543461
// MI455X (gfx1250) — compile-verified
//
#include <hip/hip_runtime.h>
#include <stdint.h>

typedef float v2f __attribute__((ext_vector_type(2)));
typedef float v8f __attribute__((ext_vector_type(8)));

#define B_TOT   8192
#define NFLD    32
#define DIM     256
#define NPAIR   496            // 32*31/2
#define CHUNK   32             // K columns per LDS chunk
#define KSTEPS  (CHUNK / 4)    // 8 wmma k-steps per chunk
#define NCHUNK  (DIM / CHUNK)  // 8 chunks
#define LSTRIDE 36             // padded row stride (dwords), = 4*9, 9 odd -> conflict-free b64 frag loads
#define WAVES   8
#define LDSBUF  (NFLD * LSTRIDE)   // 1152 dwords per chunk buffer

// Direct memory -> LDS async copy (16B per lane), tracked by ASYNCcnt.
// VDST = per-lane LDS byte address (low 32 bits of generic pointer), VADDR = 64-bit global address.
__device__ __forceinline__ void async_ld_b128(const float* g, const float* l) {
    unsigned           lds32 = (unsigned)(uintptr_t)l;
    unsigned long long ga    = (unsigned long long)(uintptr_t)g;
    asm volatile("global_load_async_to_lds_b128 %0, %1, off"
                 :: "v"(lds32), "v"(ga) : "memory");
}

__global__ __launch_bounds__(WAVES * 32)
void dlrm_interact_wmma(const float* __restrict__ x, float* __restrict__ out) {
    // 8 waves * 2 buffers * 1152 dwords * 4B = 73,728 B -> 4 workgroups per WGP (320 KB LDS)
    __shared__ float smem[WAVES * 2 * LDSBUF];

    const int lane  = threadIdx.x & 31;
    const int wave  = threadIdx.x >> 5;
    const int batch = blockIdx.x * WAVES + wave;

    float* lds = &smem[wave * 2 * LDSBUF];
    const float* xb = x + (size_t)batch * (NFLD * DIM);

    // Async-copy mapping: each step moves 4 rows; 8 lanes/row, 16B (b128) per lane -> 512B/step, 128B contiguous per row.
    const int aRowSub = lane >> 3;          // 0..3: which of the 4 rows in this step
    const int aCol    = (lane & 7) * 4;     // dword column within chunk (8 lanes * 4 dwords = 32)

    // Fragment mapping (identical for A-role and B-role of wmma_f32_16x16x4):
    // lane L: row = base + (L&15), k-offset = 2*(L>>4); ds_load_b64 -> {X[row][k], X[row][k+1]}
    const int fRow = lane & 15;
    const int fK   = (lane >> 4) * 2;

    v8f c00 = {};   // rows 0-15  x cols 0-15
    v8f c01 = {};   // rows 0-15  x cols 16-31
    v8f c11 = {};   // rows 16-31 x cols 16-31

    // Prologue: async-copy chunk 0 into buffer 0.
#pragma unroll
    for (int s = 0; s < 8; ++s) {
        const int row = 4 * s + aRowSub;
        async_ld_b128(xb + row * DIM + aCol, lds + row * LSTRIDE + aCol);
    }

#pragma unroll
    for (int c = 0; c < NCHUNK; ++c) {
        const float* buf = lds + (c & 1) * LDSBUF;

        if (c + 1 < NCHUNK) {
            // Issue next chunk's async copies into the other buffer, then wait
            // for *this* chunk's 8 copies (in-order completion) to land.
            float*       nbuf = lds + ((c + 1) & 1) * LDSBUF;
            const float* xc   = xb + (c + 1) * CHUNK;
#pragma unroll
            for (int s = 0; s < 8; ++s) {
                const int row = 4 * s + aRowSub;
                async_ld_b128(xc + row * DIM + aCol, nbuf + row * LSTRIDE + aCol);
            }
            asm volatile("s_wait_asynccnt 8" ::: "memory");
        } else {
            asm volatile("s_wait_asynccnt 0" ::: "memory");
        }

        // Compute: per k-step, 2 conflict-free ds_load_b64 feed 3 f32 WMMAs.
#pragma unroll
        for (int k = 0; k < KSTEPS; ++k) {
            const v2f aLo = *(const v2f*)(buf + fRow        * LSTRIDE + 4 * k + fK);
            const v2f aHi = *(const v2f*)(buf + (fRow + 16) * LSTRIDE + 4 * k + fK);
            c00 = __builtin_amdgcn_wmma_f32_16x16x4_f32(false, aLo, false, aLo,
                                                        (short)0, c00, false, false);
            c01 = __builtin_amdgcn_wmma_f32_16x16x4_f32(false, aLo, false, aHi,
                                                        (short)0, c01, false, false);
            c11 = __builtin_amdgcn_wmma_f32_16x16x4_f32(false, aHi, false, aHi,
                                                        (short)0, c11, false, false);
        }
    }

    // Epilogue: scatter strictly-upper-triangular entries (row-major pair order).
    // C layout: vgpr r, lanes 0-15 -> (M=r, N=lane); lanes 16-31 -> (M=r+8, N=lane-16)
    float* ob = out + (size_t)batch * NPAIR;
    const int n0    = lane & 15;
    const int mBase = (lane >> 4) * 8;

#pragma unroll
    for (int r = 0; r < 8; ++r) {
        const int m = r + mBase;

        if (n0 > m) {   // tile (0,0): i=m, j=n0
            const int i = m, j = n0;
            ob[i * (2 * NFLD - i - 1) / 2 + (j - i - 1)] = c00[r];
        }
        {               // tile (0,1): i=m, j=n0+16 (always strictly upper)
            const int i = m, j = n0 + 16;
            ob[i * (2 * NFLD - i - 1) / 2 + (j - i - 1)] = c01[r];
        }
        if (n0 > m) {   // tile (1,1): i=m+16, j=n0+16
            const int i = m + 16, j = n0 + 16;
            ob[i * (2 * NFLD - i - 1) / 2 + (j - i - 1)] = c11[r];
        }
    }
}

extern "C" void kernel_launch(void* const* d_in, const int* in_sizes, int n_in,
                              void* d_out, int out_size, void* d_ws, size_t ws_size,
                              hipStream_t stream) {
    (void)in_sizes; (void)n_in; (void)out_size; (void)d_ws; (void)ws_size;
    const float* x = (const float*)d_in[0];
    float* out     = (float*)d_out;

    dim3 grid(B_TOT / WAVES);   // 1024 workgroups, one batch per wave
    dim3 block(WAVES * 32);     // 8 wave32s
    hipLaunchKernelGGL(dlrm_interact_wmma, grid, block, 0, stream, x, out);
}